// Model_36713380446750
// MI455X (gfx1250) — compile-verified
//
#include <hip/hip_runtime.h>

typedef __attribute__((ext_vector_type(2))) float v2f;
typedef __attribute__((ext_vector_type(8))) float v8f;

#define NN   4096   // N*N
#define KDIM 16
#define BB   64
#define PP   128
#define NMAT 64

// ---------------------------------------------------------------------------
// Kernel 1: wg = (weights^2) @ g   [16 x 4096]  via V_WMMA_F32_16X16X4_F32
// Also emits ET[(k*64+j)*64 + i] = exp(-wg[k][i*64+j])   (transposed exp table)
// Grid: 256 waves, one 16-wide column tile of wg each; K loop = 1024 steps.
// ---------------------------------------------------------------------------
__global__ __launch_bounds__(32) void k_wg(const float* __restrict__ weights,
                                           const float* __restrict__ g,
                                           float* __restrict__ wg,
                                           float* __restrict__ ET) {
    const int n0   = blockIdx.x * 16;
    const int lane = threadIdx.x;
    const int half = lane >> 4;    // 0: K=k0,k0+1  1: K=k0+2,k0+3
    const int l16  = lane & 15;
    v8f acc = {0.f, 0.f, 0.f, 0.f, 0.f, 0.f, 0.f, 0.f};
    for (int k0 = 0; k0 < NN; k0 += 4) {
        const int ka = k0 + 2 * half;
        float a0 = weights[l16 * NN + ka];
        float a1 = weights[l16 * NN + ka + 1];
        a0 *= a0;
        a1 *= a1;
        const float b0 = g[(size_t)ka * NN + n0 + l16];
        const float b1 = g[(size_t)(ka + 1) * NN + n0 + l16];
        v2f A  = {a0, a1};
        v2f Bv = {b0, b1};
        acc = __builtin_amdgcn_wmma_f32_16x16x4_f32(false, A, false, Bv,
                                                    (short)0, acc, false, false);
    }
#pragma unroll
    for (int r = 0; r < 8; ++r) {
        const int krow = r + 8 * half;   // 0..15
        const int col  = n0 + l16;       // 0..4095
        const float v  = acc[r];
        wg[krow * NN + col] = v;
        const int j = col & 63;
        const int i = col >> 6;
        ET[(krow * 64 + j) * 64 + i] = expf(-v);
    }
}

// ---------------------------------------------------------------------------
// Kernel 2 (bandwidth kernel): w_out[b,p,:] = wg[idx[b,p], :]
// idx recovered from the one-hot mask row (16 floats, one cacheline, L2-hot).
// 8192 blocks x 256 threads, float4 stores: the 134 MB that sets the runtime.
// ---------------------------------------------------------------------------
__global__ __launch_bounds__(256) void k_w(const float* __restrict__ x_mask,
                                           const float* __restrict__ wg,
                                           float* __restrict__ w_out) {
    const int bp = blockIdx.x;                       // b*128 + p
    const float* mrow = x_mask + (size_t)bp * KDIM;
    int k = 0;
#pragma unroll
    for (int t = 1; t < KDIM; ++t) k = (mrow[t] > 0.5f) ? t : k;
    const float4* src = (const float4*)(wg + (size_t)k * NN);
    float4* dst = (float4*)(w_out + (size_t)bp * NN);
#pragma unroll
    for (int i = threadIdx.x; i < NN / 4; i += 256) dst[i] = src[i];
}

// ---------------------------------------------------------------------------
// Kernel 3a: S[b,k,j] = sum_p x_mask[b,p,k] * xr[b,p,j]
// Per batch: [16 x 128] @ [128 x 64]. One wave per (b, 16-wide j tile).
// ---------------------------------------------------------------------------
__global__ __launch_bounds__(32) void k_s(const float* __restrict__ x,
                                          const float* __restrict__ x_mask,
                                          float* __restrict__ S) {
    const int b    = blockIdx.x >> 2;
    const int n0   = (blockIdx.x & 3) * 16;
    const int lane = threadIdx.x;
    const int half = lane >> 4;
    const int l16  = lane & 15;
    const float* xb = x + (size_t)b * (PP * NMAT);       // xr[b][p][j] flat view
    const float* mb = x_mask + (size_t)b * (PP * KDIM);  // mask[b][p][k]
    v8f acc = {0.f, 0.f, 0.f, 0.f, 0.f, 0.f, 0.f, 0.f};
    for (int p0 = 0; p0 < PP; p0 += 4) {
        const int pa   = p0 + 2 * half;
        const float a0 = mb[pa * KDIM + l16];            // A[m=k][kk=p]
        const float a1 = mb[(pa + 1) * KDIM + l16];
        const float b0 = xb[pa * NMAT + n0 + l16];       // B[kk=p][n=j]
        const float b1 = xb[(pa + 1) * NMAT + n0 + l16];
        v2f A  = {a0, a1};
        v2f Bv = {b0, b1};
        acc = __builtin_amdgcn_wmma_f32_16x16x4_f32(false, A, false, Bv,
                                                    (short)0, acc, false, false);
    }
#pragma unroll
    for (int r = 0; r < 8; ++r) {
        const int k = r + 8 * half;
        S[(size_t)b * 1024 + k * 64 + n0 + l16] = acc[r];
    }
}

// ---------------------------------------------------------------------------
// Kernel 3b: Z[64 x 64] = S_flat[64 x 1024] @ ET[1024 x 64]
// 16 waves, one 16x16 tile each; K loop = 256 steps.
// ---------------------------------------------------------------------------
__global__ __launch_bounds__(32) void k_z(const float* __restrict__ S,
                                          const float* __restrict__ ET,
                                          float* __restrict__ Z) {
    const int m0   = (blockIdx.x >> 2) * 16;
    const int n0   = (blockIdx.x & 3) * 16;
    const int lane = threadIdx.x;
    const int half = lane >> 4;
    const int l16  = lane & 15;
    v8f acc = {0.f, 0.f, 0.f, 0.f, 0.f, 0.f, 0.f, 0.f};
    for (int k0 = 0; k0 < 1024; k0 += 4) {
        const int ka   = k0 + 2 * half;
        const float a0 = S[(m0 + l16) * 1024 + ka];
        const float a1 = S[(m0 + l16) * 1024 + ka + 1];
        const float b0 = ET[ka * 64 + n0 + l16];
        const float b1 = ET[(ka + 1) * 64 + n0 + l16];
        v2f A  = {a0, a1};
        v2f Bv = {b0, b1};
        acc = __builtin_amdgcn_wmma_f32_16x16x4_f32(false, A, false, Bv,
                                                    (short)0, acc, false, false);
    }
#pragma unroll
    for (int r = 0; r < 8; ++r) {
        Z[(m0 + r + 8 * half) * 64 + n0 + l16] = acc[r];
    }
}

// ---------------------------------------------------------------------------
// Launch. Output layout: [ Z (4096 floats) | w (33,554,432 floats) ].
// Workspace: wg (64K floats) | ET (64K floats) | S (64K floats) = 768 KB.
// ---------------------------------------------------------------------------
extern "C" void kernel_launch(void* const* d_in, const int* in_sizes, int n_in,
                              void* d_out, int out_size, void* d_ws, size_t ws_size,
                              hipStream_t stream) {
    const float* x       = (const float*)d_in[0];   // [64, 64, 128]
    const float* x_mask  = (const float*)d_in[1];   // [64, 128, 16]
    const float* weights = (const float*)d_in[2];   // [16, 4096]
    const float* g       = (const float*)d_in[3];   // [4096, 4096]

    float* out   = (float*)d_out;
    float* Z     = out;                  // 64*64
    float* w_out = out + 64 * 64;        // 64*128*4096

    float* ws = (float*)d_ws;
    float* wg = ws;                      // 16*4096
    float* ET = ws + 65536;              // 1024*64
    float* S  = ws + 131072;             // 64*16*64

    k_wg<<<NN / 16, 32, 0, stream>>>(weights, g, wg, ET);
    k_s<<<BB * 4, 32, 0, stream>>>(x, x_mask, S);
    k_w<<<BB * PP, 256, 0, stream>>>(x_mask, wg, w_out);
    k_z<<<16, 32, 0, stream>>>(S, ET, Z);
}